// MDGAT_15367392985271
// MI455X (gfx1250) — compile-verified
//
#include <hip/hip_runtime.h>
#include <hip/hip_bf16.h>

// Problem constants (match reference)
#define B_SZ    8
#define D_SZ    128
#define H_SZ    4
#define DH_SZ   32
#define N_SZ    2048
#define M_SZ    2048
#define TOPK_SZ 32

typedef float v2f __attribute__((ext_vector_type(2)));
typedef float v8f __attribute__((ext_vector_type(8)));

// D = A(16x4 f32) * B(4x16 f32) + C(16x16 f32), wave32 WMMA
__device__ __forceinline__ v8f wmma_f32_k4(v2f a, v2f b, v8f c) {
    return __builtin_amdgcn_wmma_f32_16x16x4_f32(false, a, false, b,
                                                 (short)0, c, false, false);
}

// ---------------------------------------------------------------------------
// f32 WMMA GEMM, fully unrolled over compile-time IN:
//   Y[bz, o, n] = W[o, :] @ X[bz, :, n] + bias[o]
// X is the channel-concat of X0 (channels [0,SPLIT)) and X1 (rest).
// BN / RELU epilogue selected at compile time.
// Block = 128 threads (4 wave32); each wave owns one 16x16 tile (block: 16x64).
// All inner-loop load offsets are compile-time immediates off one base pointer.
// ---------------------------------------------------------------------------
template<int IN, int SPLIT, bool BN, bool RELU>
__global__ __launch_bounds__(128)
void gemm_kernel(const float* __restrict__ W, const float* __restrict__ bias,
                 const float* __restrict__ X0, const float* __restrict__ X1,
                 float* __restrict__ Y, int OUT, int NC,
                 const float* __restrict__ g,  const float* __restrict__ bet,
                 const float* __restrict__ mu, const float* __restrict__ var)
{
    const int lane  = threadIdx.x & 31;
    const int wave  = threadIdx.x >> 5;
    const int bz    = blockIdx.z;
    const int o0    = blockIdx.x * 16;
    const int n0    = (blockIdx.y * 4 + wave) * 16;
    const int mrow  = lane & 15;                 // A's M index == B's N index
    const int khalf = (lane < 16) ? 0 : 2;       // K sub-offset per ISA layout
    const int rbase = (lane < 16) ? 0 : 8;

    // Per-lane base pointers; all k offsets below are compile-time constants.
    const float* wp = W + (size_t)(o0 + mrow) * IN + khalf;            // v2f @ wp+k
    const float* x0 = X0 + ((size_t)bz * SPLIT + khalf) * NC + n0 + mrow;
    const float* x1 = nullptr;
    if constexpr (IN > SPLIT)
        x1 = X1 + ((size_t)bz * (IN - SPLIT) + khalf) * NC + n0 + mrow;

    v8f acc = {};
#pragma unroll
    for (int k = 0; k < IN; k += 4) {
        v2f a = *(const v2f*)(wp + k);           // W[o, k+khalf .. k+khalf+1]
        v2f b;
        if (k < SPLIT) {
            b.x = x0[(size_t)k       * NC];
            b.y = x0[(size_t)(k + 1) * NC];
        } else {
            b.x = x1[(size_t)(k - SPLIT)     * NC];
            b.y = x1[(size_t)(k - SPLIT + 1) * NC];
        }
        acc = wmma_f32_k4(a, b, acc);
    }

    // Epilogue: bias (+ BN + ReLU); 8 rows per lane, constant store offsets.
    float* yp = Y + ((size_t)bz * OUT + o0 + rbase) * NC + n0 + mrow;
#pragma unroll
    for (int r = 0; r < 8; ++r) {
        const int orow = o0 + rbase + r;
        float val = acc[r] + bias[orow];
        if constexpr (BN) {
            val = (val - mu[orow]) * rsqrtf(var[orow] + 1e-5f);
            val = val * g[orow] + bet[orow];
        }
        if constexpr (RELU) val = fmaxf(val, 0.0f);
        yp[(size_t)r * NC] = val;
    }
}

// ---------------------------------------------------------------------------
// Attention: per block = (16 query rows, one head h, one batch b).
// Phase 1: scores[16, M] = Q_h^T K_h / sqrt(DH) into LDS via WMMA.
//          Q fragments hoisted into 8 v2f registers (loop-invariant);
//          K fragments streamed with constant offsets + prefetch.
// Phase 2: per-row top-32 (one wave per row), lane-resident softmax.
// Phase 3: msg[dh, n] = sum_i S_i * V_h[dh, idx_i]  (lane = dh).
// Head-interleaved channel layout: c = dh*H + h  (reference .reshape order).
// ---------------------------------------------------------------------------
__global__ __launch_bounds__(128)
void attn_topk_kernel(const float* __restrict__ q, const float* __restrict__ kk,
                      const float* __restrict__ v, float* __restrict__ msg)
{
    extern __shared__ float sc[];   // 16 * M_SZ floats = 128 KB

    const int lane  = threadIdx.x & 31;
    const int wave  = threadIdx.x >> 5;
    const int h     = blockIdx.y;
    const int bz    = blockIdx.z;
    const int n0    = blockIdx.x * 16;
    const int mrow  = lane & 15;
    const int khalf = (lane < 16) ? 0 : 2;
    const int rbase = (lane < 16) ? 0 : 8;

    const float* qb = q  + (size_t)bz * D_SZ * N_SZ;
    const float* kb = kk + (size_t)bz * D_SZ * M_SZ;
    const float* vb = v  + (size_t)bz * D_SZ * M_SZ;

    // ---- Phase 1: dense score stripe via WMMA ----
    // A fragments (Q tile, 16 rows x 32 dh): loop-invariant -> registers.
    // channel c = dh*H + h  =>  stride per dh is H*N (resp. H*M).
    const float* qbase = qb + ((size_t)h + (size_t)khalf * H_SZ) * N_SZ + n0 + mrow;
    v2f afrag[DH_SZ / 4];
#pragma unroll
    for (int d4 = 0; d4 < DH_SZ / 4; ++d4) {
        afrag[d4].x = qbase[(size_t)(d4 * 4)     * H_SZ * N_SZ];
        afrag[d4].y = qbase[(size_t)(d4 * 4 + 1) * H_SZ * N_SZ];
    }

    const float* kbase0 = kb + ((size_t)h + (size_t)khalf * H_SZ) * M_SZ + mrow;
    const float scale = 0.17677669529663687f;   // 1/sqrt(32)
    for (int mt = wave; mt < M_SZ / 16; mt += 4) {
        const int m0 = mt * 16;
        const float* kbase = kbase0 + m0;
        __builtin_prefetch(kbase + 64, 0, 0);   // this wave's next K tile
        v8f acc = {};
#pragma unroll
        for (int d4 = 0; d4 < DH_SZ / 4; ++d4) {
            v2f b;
            b.x = kbase[(size_t)(d4 * 4)     * H_SZ * M_SZ];
            b.y = kbase[(size_t)(d4 * 4 + 1) * H_SZ * M_SZ];
            acc = wmma_f32_k4(afrag[d4], b, acc);
        }
#pragma unroll
        for (int r = 0; r < 8; ++r)
            sc[(size_t)(r + rbase) * M_SZ + m0 + mrow] = acc[r] * scale;
    }
    __syncthreads();

    // ---- Phase 2+3: one wave per query row ----
    for (int row = wave; row < 16; row += 4) {
        volatile float* srow = sc + (size_t)row * M_SZ;

        float myv = -INFINITY;   // lane i holds winner of iteration i
        int   myi = 0;
        for (int it = 0; it < TOPK_SZ; ++it) {
            float bv = -INFINITY;
            int   bi = 0;
            for (int j = lane; j < M_SZ; j += 32) {
                float x = srow[j];
                if (x > bv) { bv = x; bi = j; }
            }
            for (int off = 16; off; off >>= 1) {
                float ov = __shfl_xor(bv, off, 32);
                int   oi = __shfl_xor(bi, off, 32);
                if (ov > bv || (ov == bv && oi < bi)) { bv = ov; bi = oi; }
            }
            if (lane == it) { myv = bv; myi = bi; }
            if (lane == 0) srow[bi] = -INFINITY;   // same-wave LDS ops are in-order
        }

        // softmax across lanes (values descending => max is lane 0's)
        float mx = __shfl(myv, 0, 32);
        float e  = __expf(myv - mx);
        float s  = e;
        for (int off = 16; off; off >>= 1) s += __shfl_xor(s, off, 32);
        float S = e / s;

        // sparse gather-matmul: lane = dh
        float accm = 0.0f;
        for (int i = 0; i < TOPK_SZ; ++i) {
            float Si = __shfl(S, i, 32);
            int   id = __shfl(myi, i, 32);
            accm += Si * vb[(size_t)(lane * H_SZ + h) * M_SZ + id];
        }
        msg[((size_t)bz * D_SZ + lane * H_SZ + h) * N_SZ + n0 + row] = accm;
    }
}

// ---------------------------------------------------------------------------
extern "C" void kernel_launch(void* const* d_in, const int* in_sizes, int n_in,
                              void* d_out, int out_size, void* d_ws, size_t ws_size,
                              hipStream_t stream)
{
    const float* x      = (const float*)d_in[0];
    const float* source = (const float*)d_in[1];
    const float* Wq = (const float*)d_in[2];  const float* bq = (const float*)d_in[3];
    const float* Wk = (const float*)d_in[4];  const float* bk = (const float*)d_in[5];
    const float* Wv = (const float*)d_in[6];  const float* bv = (const float*)d_in[7];
    const float* Wm = (const float*)d_in[8];  const float* bm = (const float*)d_in[9];
    const float* W1 = (const float*)d_in[10]; const float* b1 = (const float*)d_in[11];
    const float* g1 = (const float*)d_in[12]; const float* beta1 = (const float*)d_in[13];
    const float* mu1 = (const float*)d_in[14]; const float* var1 = (const float*)d_in[15];
    const float* W2 = (const float*)d_in[16]; const float* b2 = (const float*)d_in[17];

    float* out = (float*)d_out;

    // Workspace carve-up (floats): q,k,v,msg,msgm (2M each) + z (4M) = 14M floats
    float* ws    = (float*)d_ws;
    const size_t PL = (size_t)B_SZ * D_SZ * N_SZ;   // 2M elements
    float* qb   = ws;
    float* kbuf = qb   + PL;
    float* vbuf = kbuf + PL;
    float* msg  = vbuf + PL;
    float* msgm = msg  + PL;
    float* z    = msgm + PL;                         // B * 2D * N = 4M elements

    dim3 blk(128);

    // q/k/v projections (WMMA GEMMs, IN=128)
    gemm_kernel<D_SZ, D_SZ, false, false>
        <<<dim3(D_SZ/16, N_SZ/64, B_SZ), blk, 0, stream>>>(
        Wq, bq, x, nullptr, qb, D_SZ, N_SZ, nullptr, nullptr, nullptr, nullptr);
    gemm_kernel<D_SZ, D_SZ, false, false>
        <<<dim3(D_SZ/16, M_SZ/64, B_SZ), blk, 0, stream>>>(
        Wk, bk, source, nullptr, kbuf, D_SZ, M_SZ, nullptr, nullptr, nullptr, nullptr);
    gemm_kernel<D_SZ, D_SZ, false, false>
        <<<dim3(D_SZ/16, M_SZ/64, B_SZ), blk, 0, stream>>>(
        Wv, bv, source, nullptr, vbuf, D_SZ, M_SZ, nullptr, nullptr, nullptr, nullptr);

    // sparse attention (scores stripe in LDS, top-32, softmax, gather)
    const size_t lds_bytes = (size_t)16 * M_SZ * sizeof(float);   // 128 KB / WG
    hipFuncSetAttribute(reinterpret_cast<const void*>(attn_topk_kernel),
                        hipFuncAttributeMaxDynamicSharedMemorySize, (int)lds_bytes);
    attn_topk_kernel<<<dim3(N_SZ/16, H_SZ, B_SZ), blk, lds_bytes, stream>>>(
        qb, kbuf, vbuf, msg);

    // merge conv (IN=128)
    gemm_kernel<D_SZ, D_SZ, false, false>
        <<<dim3(D_SZ/16, N_SZ/64, B_SZ), blk, 0, stream>>>(
        Wm, bm, msg, nullptr, msgm, D_SZ, N_SZ, nullptr, nullptr, nullptr, nullptr);

    // MLP layer 1 on concat([x, msgm]) with BN + ReLU (IN=256, SPLIT=128)
    gemm_kernel<2*D_SZ, D_SZ, true, true>
        <<<dim3(2*D_SZ/16, N_SZ/64, B_SZ), blk, 0, stream>>>(
        W1, b1, x, msgm, z, 2*D_SZ, N_SZ, g1, beta1, mu1, var1);

    // MLP layer 2 -> delta (IN=256, no concat)
    gemm_kernel<2*D_SZ, 2*D_SZ, false, false>
        <<<dim3(D_SZ/16, N_SZ/64, B_SZ), blk, 0, stream>>>(
        W2, b2, z, nullptr, out, D_SZ, N_SZ, nullptr, nullptr, nullptr, nullptr);
}